// SetOfSetGlobalFeatureUpdate_33088428049081
// MI455X (gfx1250) — compile-verified
//
#include <hip/hip_runtime.h>

typedef __attribute__((ext_vector_type(2))) float v2f;
typedef __attribute__((ext_vector_type(8))) float v8f;

#define D_IN   128
#define N_VIEWS 256
#define N_POINTS 65536
#define LWS 144   // LDS stride pad: 2*144 % 64 == 32 -> lane halves hit disjoint banks

// ---------------------------------------------------------------------------
// Kernel 1: zero the accumulator regions (d_out is poisoned before timing)
// ---------------------------------------------------------------------------
__global__ void zero_kernel(float* __restrict__ out, size_t n_out,
                            float* __restrict__ ws, size_t n_ws) {
    size_t i = (size_t)blockIdx.x * blockDim.x + threadIdx.x;
    size_t stride = (size_t)gridDim.x * blockDim.x;
    for (size_t j = i; j < n_out; j += stride) out[j] = 0.0f;
    for (size_t j = i; j < n_ws; j += stride) ws[j] = 0.0f;
}

// ---------------------------------------------------------------------------
// Kernel 2: scatter pass over nnz rows.
//   - col_sum: direct global f32 atomics (65536 segments, low contention,
//     32MB accumulator lives in the 192MB L2)
//   - row_sum: 256x128 f32 partial table in LDS (ds_add_f32), flushed once
//   - global sum: per-lane registers -> LDS -> one atomic per column/block
// One wave processes one nnz row per iteration: lane l owns features 4l..4l+3.
// ---------------------------------------------------------------------------
__global__ __launch_bounds__(256, 2)
void scatter_kernel(const float* __restrict__ values,
                    const int* __restrict__ row_idx,
                    const int* __restrict__ col_idx,
                    float* __restrict__ col_sum,   // [N_POINTS][128] (d_out)
                    float* __restrict__ col_cnt,   // [N_POINTS]      (ws)
                    float* __restrict__ row_sum,   // [N_VIEWS][128]  (d_out)
                    float* __restrict__ row_cnt,   // [N_VIEWS]       (ws)
                    float* __restrict__ glob,      // [128]           (ws)
                    int nnz) {
    __shared__ float lrow[N_VIEWS * D_IN];   // 128 KB
    __shared__ float lcnt[N_VIEWS];
    __shared__ float gtab[256 * 4];

    const int tid = threadIdx.x;
    for (int j = tid; j < N_VIEWS * D_IN; j += 256) lrow[j] = 0.0f;
    lcnt[tid] = 0.0f;
    __syncthreads();

    const int lane = tid & 31;
    const int gw = blockIdx.x * 8 + (tid >> 5);
    const int nw = gridDim.x * 8;
    const int co = lane * 4;

    float g0 = 0.f, g1 = 0.f, g2 = 0.f, g3 = 0.f;

    const float4* vp = reinterpret_cast<const float4*>(values);
    for (int i = gw; i < nnz; i += nw) {
        const int r = row_idx[i];
        const int c = col_idx[i];
        const float4 v = vp[(size_t)i * 32 + lane];

        float* cs = col_sum + (size_t)c * D_IN + co;
        unsafeAtomicAdd(cs + 0, v.x);
        unsafeAtomicAdd(cs + 1, v.y);
        unsafeAtomicAdd(cs + 2, v.z);
        unsafeAtomicAdd(cs + 3, v.w);

        float* lr = lrow + r * D_IN + co;
        atomicAdd(lr + 0, v.x);
        atomicAdd(lr + 1, v.y);
        atomicAdd(lr + 2, v.z);
        atomicAdd(lr + 3, v.w);

        g0 += v.x; g1 += v.y; g2 += v.z; g3 += v.w;

        if (lane == 0) {
            unsafeAtomicAdd(col_cnt + c, 1.0f);
            atomicAdd(&lcnt[r], 1.0f);
        }
    }

    gtab[tid * 4 + 0] = g0;
    gtab[tid * 4 + 1] = g1;
    gtab[tid * 4 + 2] = g2;
    gtab[tid * 4 + 3] = g3;
    __syncthreads();

    // flush row partials
    for (int j = tid; j < N_VIEWS * D_IN; j += 256) {
        float v = lrow[j];
        if (v != 0.0f) unsafeAtomicAdd(row_sum + j, v);
    }
    if (lcnt[tid] != 0.0f) unsafeAtomicAdd(row_cnt + tid, lcnt[tid]);

    // reduce global partials: column c is held by lane c/4, component c%4 of each wave
    if (tid < D_IN) {
        float s = 0.0f;
#pragma unroll
        for (int w = 0; w < 8; ++w)
            s += gtab[(w * 32 + (tid >> 2)) * 4 + (tid & 3)];
        unsafeAtomicAdd(glob + tid, s);
    }
}

// ---------------------------------------------------------------------------
// Kernel 3: in-place fused (sum/cnt) @ W + b using V_WMMA_F32_16X16X4_F32.
// Each wave owns one 16-row strip of io: loads all 32 A fragments (K=128)
// scaled by 1/max(cnt,1), then for each of 8 N-tiles chains 32 WMMA ops and
// stores the 16x16 result. In-place is safe: a strip is read fully into
// registers before any store, and strips are wave-exclusive.
//
// f32 WMMA layouts (ISA 7.12.2):
//   A 16x4 : lane(0-15)->M=lane, v0=K0,v1=K1 ; lane(16-31)-> v0=K2,v1=K3
//   B 4x16 : lane&15 -> N ;      v0=K0|K2, v1=K1|K3 across lane halves
//   C/D    : lane&15 -> N ; VGPR j -> M = j + 8*(lane>>4)
// ---------------------------------------------------------------------------
__global__ __launch_bounds__(256, 1)
void mean_gemm_kernel(float* __restrict__ io,       // [mtiles*16][128] in/out
                      const float* __restrict__ cnt,
                      const float* __restrict__ Wm, // [128][128] row-major
                      const float* __restrict__ bias,
                      int mtiles) {
    __shared__ float lw[D_IN * LWS];
    __shared__ float lb[D_IN];

    const int tid = threadIdx.x;
    for (int j = tid; j < D_IN * D_IN; j += 256) {
        int k = j >> 7, n = j & 127;
        lw[k * LWS + n] = Wm[j];
    }
    if (tid < D_IN) lb[tid] = bias[tid];
    __syncthreads();

    const int wave = tid >> 5;
    const int lane = tid & 31;
    const int half = lane >> 4;     // which K-half this lane holds
    const int lx   = lane & 15;     // M for A-frag, N for B/C-frags
    const int mt   = blockIdx.x * 8 + wave;
    if (mt >= mtiles) return;       // wave-uniform; EXEC stays all-ones

    const int m = mt * 16 + lx;
    const float inv = 1.0f / fmaxf(cnt[m], 1.0f);

    // Load + scale all A fragments for this strip (K = 0..127)
    const v2f* arow = reinterpret_cast<const v2f*>(io + (size_t)m * D_IN);
    v2f afrag[32];
#pragma unroll
    for (int kb = 0; kb < 32; ++kb) {
        v2f t = arow[kb * 2 + half];          // floats [4kb+2h, 4kb+2h+1]
        afrag[kb] = t * inv;
    }

    for (int nt = 0; nt < 8; ++nt) {
        const int n0 = nt * 16;
        const float bv = lb[n0 + lx];
        v8f c;
#pragma unroll
        for (int j = 0; j < 8; ++j) c[j] = bv;  // bias depends on N only

#pragma unroll
        for (int kb = 0; kb < 32; ++kb) {
            const int k0 = kb * 4 + half * 2;
            v2f bfrag;
            bfrag.x = lw[k0 * LWS + n0 + lx];
            bfrag.y = lw[(k0 + 1) * LWS + n0 + lx];
            c = __builtin_amdgcn_wmma_f32_16x16x4_f32(
                    false, afrag[kb], false, bfrag, (short)0, c, false, false);
        }

#pragma unroll
        for (int j = 0; j < 8; ++j)
            io[(size_t)(mt * 16 + j + 8 * half) * D_IN + n0 + lx] = c[j];
    }
}

// ---------------------------------------------------------------------------
// Kernel 4: global features: (sum/nnz) @ W_glob + b_glob  -> [1][128]
// ---------------------------------------------------------------------------
__global__ void global_feat_kernel(const float* __restrict__ gsum,
                                   const float* __restrict__ Wg,
                                   const float* __restrict__ bg,
                                   float* __restrict__ out, float invn) {
    const int c = threadIdx.x;  // 128 threads
    float acc = bg[c];
#pragma unroll 4
    for (int k = 0; k < D_IN; ++k)
        acc = fmaf(gsum[k] * invn, Wg[k * D_IN + c], acc);
    out[c] = acc;
}

// ---------------------------------------------------------------------------
extern "C" void kernel_launch(void* const* d_in, const int* in_sizes, int n_in,
                              void* d_out, int out_size, void* d_ws, size_t ws_size,
                              hipStream_t stream) {
    const float* values  = (const float*)d_in[0];
    const int*   row_idx = (const int*)d_in[1];
    const int*   col_idx = (const int*)d_in[2];
    const float* W_sp    = (const float*)d_in[5];
    const float* b_sp    = (const float*)d_in[6];
    const float* W_view  = (const float*)d_in[7];
    const float* b_view  = (const float*)d_in[8];
    const float* W_glob  = (const float*)d_in[9];
    const float* b_glob  = (const float*)d_in[10];

    const int nnz = in_sizes[0] / D_IN;

    float* out      = (float*)d_out;
    float* col_acc  = out;                                   // [65536][128]
    float* view_acc = out + (size_t)N_POINTS * D_IN;         // [256][128]
    float* glob_out = view_acc + (size_t)N_VIEWS * D_IN;     // [128]

    float* ws       = (float*)d_ws;
    float* col_cnt  = ws;                          // 65536
    float* row_cnt  = ws + N_POINTS;               // 256
    float* glob_sum = ws + N_POINTS + N_VIEWS;     // 128
    const size_t n_ws_zero = (size_t)N_POINTS + N_VIEWS + D_IN;

    zero_kernel<<<2048, 256, 0, stream>>>(out, (size_t)out_size, ws, n_ws_zero);

    scatter_kernel<<<1024, 256, 0, stream>>>(values, row_idx, col_idx,
                                             col_acc, col_cnt,
                                             view_acc, row_cnt,
                                             glob_sum, nnz);

    // scenepoint_features: 65536/16 = 4096 m-tiles, 8 per block
    mean_gemm_kernel<<<512, 256, 0, stream>>>(col_acc, col_cnt, W_sp, b_sp,
                                              N_POINTS / 16);
    // view_features: 256/16 = 16 m-tiles
    mean_gemm_kernel<<<2, 256, 0, stream>>>(view_acc, row_cnt, W_view, b_view,
                                            N_VIEWS / 16);

    global_feat_kernel<<<1, 128, 0, stream>>>(glob_sum, W_glob, b_glob,
                                              glob_out, 1.0f / (float)nnz);
}